// SarvamMoeTopkRouter_9010841387369
// MI455X (gfx1250) — compile-verified
//
#include <hip/hip_runtime.h>
#include <hip/hip_bf16.h>

// ---------------- problem constants (match reference) ----------------
#define HIDDEN        7168
#define NUM_EXPERTS   256
#define TOP_K         8
#define N_GROUP       8
#define TOPK_GROUP    4
#define PER_GROUP     (NUM_EXPERTS / N_GROUP)   // 32
#define SCALE_FACTOR  2.5f

// ---------------- GEMM tiling ----------------
#define M_BLOCK  64          // tokens per workgroup
#define KC       32          // K chunk staged in LDS
#define LDPAD    4           // pad -> stride 36 floats: conflict-free b64 frag loads
#define LDR      (KC + LDPAD)
#define NCHUNK   (HIDDEN / KC)   // 224

typedef __attribute__((ext_vector_type(2))) float v2f;
typedef __attribute__((ext_vector_type(8))) float v8f;

typedef unsigned int       u32;
typedef unsigned long long u64;
typedef __attribute__((ext_vector_type(4))) unsigned int u32x4;
typedef __attribute__((ext_vector_type(8))) int          i32x8;
typedef __attribute__((ext_vector_type(4))) int          i32x4;

#ifndef __has_builtin
#define __has_builtin(x) 0
#endif
#if __has_builtin(__builtin_amdgcn_tensor_load_to_lds)
#define HAVE_TDM 1
#else
#define HAVE_TDM 0
#endif

#if HAVE_TDM
// ----------------------------------------------------------------------
// TDM 2-D tile load: rows x KC fp32 tile from global (row stride HIDDEN
// elements) into LDS with +4-DWORD row padding (LDS row stride = LDR=36).
// Descriptor layout per CDNA5 ISA.
// ----------------------------------------------------------------------
__device__ __forceinline__
void tdm_load_tile(u32 lds_off, u64 gaddr, u32 rows)
{
    u32x4 g0;
    g0.x = 1u;                                   // count=1, user desc, no gather
    g0.y = lds_off;                              // lds_addr (bytes)
    g0.z = (u32)(gaddr & 0xFFFFFFFFu);           // global_addr[31:0]
    g0.w = (u32)((gaddr >> 32) & 0x01FFFFFFu)    // global_addr[56:32]
         | (2u << 30);                           // type = 2 ("image")

    i32x8 g1;
    // data_size=2 (4B) | pad_enable | pad_interval=4 (32 DW) | pad_amount=3 (4 DW)
    g1[0] = (int)((2u << 16) | (1u << 20) | (4u << 22) | (3u << 25));
    g1[1] = (int)(((u32)HIDDEN & 0xFFFFu) << 16);            // tensor_dim0 lo16 @ bits 63:48
    g1[2] = (int)(((u32)HIDDEN >> 16) | (rows << 16));       // tensor_dim0 hi16, tensor_dim1 lo16
    g1[3] = (int)((rows >> 16) | ((u32)KC << 16));           // tensor_dim1 hi16, tile_dim0=KC
    g1[4] = (int)(rows & 0xFFFFu);                           // tile_dim1=rows, tile_dim2=0
    g1[5] = (int)HIDDEN;                                     // tensor_dim0_stride[31:0]
    g1[6] = 0;                                               // stride0 hi, stride1 lo
    g1[7] = 0;                                               // stride1 hi

    i32x4 gz;  gz[0] = 0; gz[1] = 0; gz[2] = 0; gz[3] = 0;   // 2-D tile: groups 2/3 unused

#if __clang_major__ >= 23
    i32x8 gz8; gz8[0]=0; gz8[1]=0; gz8[2]=0; gz8[3]=0; gz8[4]=0; gz8[5]=0; gz8[6]=0; gz8[7]=0;
    __builtin_amdgcn_tensor_load_to_lds(g0, g1, gz, gz, gz8, 0);
#else
    __builtin_amdgcn_tensor_load_to_lds(g0, g1, gz, gz, 0);
#endif
}
#endif // HAVE_TDM

// ======================================================================
// Kernel 1: router logits GEMM, fp32 via V_WMMA_F32_16X16X4_F32
//   grid.x = T / M_BLOCK, block = 256 (8 wave32)
//   wave w: M tile = w>>1 (16 tokens), N tiles = (w&1)*8 .. +8 (8x16 experts)
//   K chunks double-buffered in LDS; chunk c+1 DMA'd (TDM) while computing c.
// ======================================================================
__global__ __launch_bounds__(256, 2)
void router_gemm_wmma(const float* __restrict__ hs,
                      const float* __restrict__ wgt,
                      float* __restrict__ logits)
{
    __shared__ float lds_a[2][M_BLOCK * LDR];        //  9 KB x2
    __shared__ float lds_b[2][NUM_EXPERTS * LDR];    // 36 KB x2  (total 90 KB < 320 KB)

    const int tid   = threadIdx.x;
    const int wave  = tid >> 5;
    const int lane  = tid & 31;
    const int half  = lane >> 4;      // 0: K 0..1 , 1: K 2..3
    const int lr    = lane & 15;      // M (A) / N (B) index inside tile

    const int block_m = blockIdx.x * M_BLOCK;
    const int m_tile  = wave >> 1;            // 0..3
    const int n_base  = (wave & 1) * 128;     // 8 tiles of 16 experts

    v8f acc[8] = {};                          // 64 VGPRs of accumulators

    auto compute_chunk = [&](int buf) {
        const float* pa = &lds_a[buf][(m_tile * 16 + lr) * LDR];
        #pragma unroll
        for (int kk = 0; kk < KC; kk += 4) {
            const int c = kk + 2 * half;                 // even -> 8B aligned
            v2f a = *(const v2f*)(pa + c);               // A frag: M=lr, K=c..c+1
            #pragma unroll
            for (int t = 0; t < 8; ++t) {
                v2f b = *(const v2f*)(&lds_b[buf][(n_base + t * 16 + lr) * LDR + c]);
                acc[t] = __builtin_amdgcn_wmma_f32_16x16x4_f32(
                    /*neg_a=*/false, a, /*neg_b=*/false, b,
                    /*c_mod=*/(short)0, acc[t],
                    /*reuse_a=*/false, /*reuse_b=*/false);
            }
        }
    };

#if HAVE_TDM
    // -------- Tensor Data Mover pipeline (wave 0 drives the DMA) --------
    const u64 gA = (u64)(size_t)(hs + (size_t)block_m * HIDDEN);
    const u64 gB = (u64)(size_t)wgt;
    const u32 ldsA[2] = { (u32)(size_t)&lds_a[0][0], (u32)(size_t)&lds_a[1][0] };
    const u32 ldsB[2] = { (u32)(size_t)&lds_b[0][0], (u32)(size_t)&lds_b[1][0] };

    if (tid < 32) {                                   // wave 0 only (TDM ignores EXEC)
        tdm_load_tile(ldsA[0], gA, M_BLOCK);
        tdm_load_tile(ldsB[0], gB, NUM_EXPERTS);
        __builtin_amdgcn_s_wait_tensorcnt(0);
    }
    __syncthreads();

    for (int c = 0; c < NCHUNK; ++c) {
        if (c + 1 < NCHUNK && tid < 32) {             // DMA chunk c+1 during compute of c
            const u64 koff = (u64)(c + 1) * (KC * 4);
            const int nb = (c + 1) & 1;
            tdm_load_tile(ldsA[nb], gA + koff, M_BLOCK);
            tdm_load_tile(ldsB[nb], gB + koff, NUM_EXPERTS);
        }
        compute_chunk(c & 1);
        if (tid < 32) __builtin_amdgcn_s_wait_tensorcnt(0);
        __syncthreads();
    }
#else
    // -------- fallback: register-staged double buffering --------
    float4 areg[2];
    float4 breg[8];

    auto load_chunk = [&](int k0) {
        #pragma unroll
        for (int j = 0; j < 2; ++j) {          // A: 64 rows x 8 float4 = 512
            int idx = tid + j * 256;
            int row = idx >> 3, seg = idx & 7;
            areg[j] = *(const float4*)(hs + (size_t)(block_m + row) * HIDDEN + k0 + seg * 4);
        }
        #pragma unroll
        for (int j = 0; j < 8; ++j) {          // B: 256 rows x 8 float4 = 2048
            int idx = tid + j * 256;
            int row = idx >> 3, seg = idx & 7;
            breg[j] = *(const float4*)(wgt + (size_t)row * HIDDEN + k0 + seg * 4);
        }
    };
    auto stage_chunk = [&](int buf) {
        #pragma unroll
        for (int j = 0; j < 2; ++j) {
            int idx = tid + j * 256;
            int row = idx >> 3, seg = idx & 7;
            *(float4*)(&lds_a[buf][row * LDR + seg * 4]) = areg[j];
        }
        #pragma unroll
        for (int j = 0; j < 8; ++j) {
            int idx = tid + j * 256;
            int row = idx >> 3, seg = idx & 7;
            *(float4*)(&lds_b[buf][row * LDR + seg * 4]) = breg[j];
        }
    };

    load_chunk(0);
    stage_chunk(0);
    __syncthreads();
    for (int c = 0; c < NCHUNK; ++c) {
        if (c + 1 < NCHUNK) load_chunk((c + 1) * KC);
        compute_chunk(c & 1);
        __syncthreads();
        if (c + 1 < NCHUNK) {
            stage_chunk((c + 1) & 1);
            __syncthreads();
        }
    }
#endif

    // epilogue: D layout — VGPR r: lanes 0-15 -> M=r, lanes 16-31 -> M=8+r; N=lr
    #pragma unroll
    for (int t = 0; t < 8; ++t) {
        const int col = n_base + t * 16 + lr;
        #pragma unroll
        for (int r = 0; r < 8; ++r) {
            const int row = block_m + m_tile * 16 + half * 8 + r;
            logits[(size_t)row * NUM_EXPERTS + col] = acc[t][r];
        }
    }
}

// ======================================================================
// Kernel 2: routing — sigmoid+bias, group top2 sums, top4 groups,
// masked top8 experts, normalized weights * 2.5. One block per token.
// ======================================================================
__global__ __launch_bounds__(256)
void router_topk(const float* __restrict__ logits,
                 const float* __restrict__ bias,
                 float* __restrict__ topk_w,
                 int*   __restrict__ topk_i)
{
    __shared__ float s_score[NUM_EXPERTS];
    __shared__ float s_corr[NUM_EXPERTS];
    __shared__ float s_gscore[N_GROUP];

    const int token = blockIdx.x;
    const int e     = threadIdx.x;

    const float lg   = logits[(size_t)token * NUM_EXPERTS + e];
    const float sc   = 1.0f / (1.0f + __expf(-lg));
    const float corr = sc + bias[e];
    s_score[e] = sc;
    s_corr[e]  = corr;
    __syncthreads();

    if (e < N_GROUP) {                         // per-group top-2 sum
        float m1 = -INFINITY, m2 = -INFINITY;
        const float* g = &s_corr[e * PER_GROUP];
        for (int i = 0; i < PER_GROUP; ++i) {
            float v = g[i];
            if (v > m1)      { m2 = m1; m1 = v; }
            else if (v > m2) { m2 = v; }
        }
        s_gscore[e] = m1 + m2;
    }
    __syncthreads();

    if (e == 0) {
        // top-4 groups (first-max tie-break, same as lax.top_k)
        bool gsel[N_GROUP];
        float gs[N_GROUP];
        for (int g = 0; g < N_GROUP; ++g) { gs[g] = s_gscore[g]; gsel[g] = false; }
        for (int k = 0; k < TOPK_GROUP; ++k) {
            int best = 0; float bv = -INFINITY;
            for (int g = 0; g < N_GROUP; ++g)
                if (!gsel[g] && gs[g] > bv) { bv = gs[g]; best = g; }
            gsel[best] = true;
        }

        // masked top-8 experts: value = gsel ? corrected : 0.0 (exact ref semantics)
        unsigned taken[NUM_EXPERTS / 32] = {0, 0, 0, 0, 0, 0, 0, 0};
        int   idx[TOP_K];
        float wv[TOP_K];
        float wsum = 0.0f;
        for (int k = 0; k < TOP_K; ++k) {
            int best = 0; float bv = -INFINITY;
            for (int x = 0; x < NUM_EXPERTS; ++x) {
                if (taken[x >> 5] & (1u << (x & 31))) continue;
                float v = gsel[x >> 5] ? s_corr[x] : 0.0f;   // group of x = x/32
                if (v > bv) { bv = v; best = x; }
            }
            taken[best >> 5] |= (1u << (best & 31));
            idx[k] = best;
            wv[k]  = s_score[best];
            wsum  += wv[k];
        }

        const float inv = SCALE_FACTOR / (wsum + 1e-20f);
        for (int k = 0; k < TOP_K; ++k) {
            topk_w[(size_t)token * TOP_K + k] = wv[k] * inv;
            topk_i[(size_t)token * TOP_K + k] = idx[k];
        }
    }
}

// ======================================================================
extern "C" void kernel_launch(void* const* d_in, const int* in_sizes, int n_in,
                              void* d_out, int out_size, void* d_ws, size_t ws_size,
                              hipStream_t stream)
{
    const float* hs   = (const float*)d_in[0];   // [T, HIDDEN]
    const float* wgt  = (const float*)d_in[1];   // [NUM_EXPERTS, HIDDEN]
    const float* bias = (const float*)d_in[2];   // [NUM_EXPERTS]

    const int T = in_sizes[0] / HIDDEN;          // 8192

    float* logits = (float*)d_out;                                   // [T, E]
    float* topk_w = logits + (size_t)T * NUM_EXPERTS;                // [T, 8]
    int*   topk_i = (int*)(topk_w + (size_t)T * TOP_K);              // [T, 8] (int bits)

    router_gemm_wmma<<<T / M_BLOCK, 256, 0, stream>>>(hs, wgt, logits);
    router_topk    <<<T,            256, 0, stream>>>(logits, bias, topk_w, topk_i);
}